// TransitionUp_23484881175029
// MI455X (gfx1250) — compile-verified
//
#include <hip/hip_runtime.h>

typedef __attribute__((ext_vector_type(2))) float v2f;
typedef __attribute__((ext_vector_type(8))) float v8f;

#define N_PTS 16384
#define M_PTS 8192
#define C_IN  256
#define C_OUT 128
#define KNN_TILE 1024

// ---------------------------------------------------------------------------
// H[M x 128] = A[M x K] @ W[K x 128] + bias   (fp32, V_WMMA_F32_16X16X4_F32)
// Wave-level tiling: each wave -> 16 rows x 64 cols (4 WMMA tiles), A fragment
// reused across the 4 column tiles. Grid is exact, no divergence (EXEC all 1s).
//
// A fragment (16x4 f32, 2 VGPR): lanes 0-15 rows 0-15 K={0,1}; lanes 16-31
// rows 0-15 K={2,3}. B fragment (4x16 f32): lanes 0-15 cols 0-15 K={0,1};
// lanes 16-31 cols 0-15 K={2,3}. D (16x16 f32, 8 VGPR): lane<16 -> M=r,
// lane>=16 -> M=r+8, N=lane&15.
// ---------------------------------------------------------------------------
__global__ __launch_bounds__(256) void gemm_bias_wmma(
    const float* __restrict__ A, const float* __restrict__ Wm,
    const float* __restrict__ bias, float* __restrict__ H,
    int M, int K)
{
  const int C = C_OUT;
  const int wave = blockIdx.x * (blockDim.x >> 5) + (threadIdx.x >> 5);
  const int lane = threadIdx.x & 31;
  const int lane15 = lane & 15;
  const int khalf = lane >> 4;                 // 0 -> K{0,1}, 1 -> K{2,3}
  const int colGroup = wave & 1;               // columns [0,64) or [64,128)
  const int rowTile = wave >> 1;               // 16-row tile

  const float* Arow = A + (size_t)(rowTile * 16 + lane15) * K + 2 * khalf;
  const int colBase = colGroup * 64 + lane15;
  const float* W0 = Wm + colBase;

  v8f acc0 = {}, acc1 = {}, acc2 = {}, acc3 = {};
  for (int kk = 0; kk < K; kk += 4) {
    v2f a = *(const v2f*)(Arow + kk);          // K pair {kk+2*khalf, +1}
    const int krow = (kk + 2 * khalf) * C;     // first K row for this half
    v2f b0, b1, b2, b3;
    b0.x = W0[krow +  0]; b0.y = W0[krow + C +  0];
    b1.x = W0[krow + 16]; b1.y = W0[krow + C + 16];
    b2.x = W0[krow + 32]; b2.y = W0[krow + C + 32];
    b3.x = W0[krow + 48]; b3.y = W0[krow + C + 48];
    acc0 = __builtin_amdgcn_wmma_f32_16x16x4_f32(false, a, false, b0, (short)0, acc0, false, false);
    acc1 = __builtin_amdgcn_wmma_f32_16x16x4_f32(false, a, false, b1, (short)0, acc1, false, false);
    acc2 = __builtin_amdgcn_wmma_f32_16x16x4_f32(false, a, false, b2, (short)0, acc2, false, false);
    acc3 = __builtin_amdgcn_wmma_f32_16x16x4_f32(false, a, false, b3, (short)0, acc3, false, false);
  }

  const float bv0 = bias[colBase +  0];
  const float bv1 = bias[colBase + 16];
  const float bv2 = bias[colBase + 32];
  const float bv3 = bias[colBase + 48];
  const int rbase = rowTile * 16 + 8 * khalf;
#pragma unroll
  for (int r = 0; r < 8; ++r) {
    const size_t off = (size_t)(rbase + r) * C + colBase;
    H[off +  0] = acc0[r] + bv0;
    H[off + 16] = acc1[r] + bv1;
    H[off + 32] = acc2[r] + bv2;
    H[off + 48] = acc3[r] + bv3;
  }
}

// ---------------------------------------------------------------------------
// Per-column mean / inv-std over M rows (biased variance, eps = 1e-5).
// One block per column.
// ---------------------------------------------------------------------------
__global__ __launch_bounds__(256) void bn_stats(
    const float* __restrict__ H, float* __restrict__ mean,
    float* __restrict__ istd, int M)
{
  const int c = blockIdx.x;
  float s = 0.f, ss = 0.f;
  for (int r = threadIdx.x; r < M; r += blockDim.x) {
    float v = H[(size_t)r * C_OUT + c];
    s += v; ss += v * v;
  }
  __shared__ float sh_s[256], sh_ss[256];
  sh_s[threadIdx.x] = s; sh_ss[threadIdx.x] = ss;
  __syncthreads();
  for (int o = 128; o > 0; o >>= 1) {
    if (threadIdx.x < o) {
      sh_s[threadIdx.x]  += sh_s[threadIdx.x + o];
      sh_ss[threadIdx.x] += sh_ss[threadIdx.x + o];
    }
    __syncthreads();
  }
  if (threadIdx.x == 0) {
    float mu  = sh_s[0] / (float)M;
    float var = sh_ss[0] / (float)M - mu * mu;
    mean[c] = mu;
    istd[c] = rsqrtf(var + 1e-5f);
  }
}

// ---------------------------------------------------------------------------
// In-place BN apply + ReLU (h_sub -> feat_sub).
// ---------------------------------------------------------------------------
__global__ __launch_bounds__(256) void bn_apply_relu(
    float* __restrict__ H, const float* __restrict__ mean,
    const float* __restrict__ istd, const float* __restrict__ g,
    const float* __restrict__ be)
{
  const int idx = blockIdx.x * blockDim.x + threadIdx.x;
  const int c = idx & (C_OUT - 1);
  float v = (H[idx] - mean[c]) * istd[c] * g[c] + be[c];
  H[idx] = v > 0.f ? v : 0.f;
}

// ---------------------------------------------------------------------------
// 3-NN over pos_sub for every query point, LDS-tiled. Emits 3 neighbor
// indices and pre-normalized inverse-square-distance weights (clamp 1e-16).
// ---------------------------------------------------------------------------
__global__ __launch_bounds__(256) void knn_top3(
    const float* __restrict__ pos, const float* __restrict__ pos_sub,
    int* __restrict__ oidx, float* __restrict__ ow)
{
  __shared__ float sx[KNN_TILE], sy[KNN_TILE], sz[KNN_TILE];
  const int i = blockIdx.x * blockDim.x + threadIdx.x;
  const float px = pos[i * 3 + 0];
  const float py = pos[i * 3 + 1];
  const float pz = pos[i * 3 + 2];

  float d0 = 3.4e38f, d1 = 3.4e38f, d2v = 3.4e38f;
  int i0 = 0, i1 = 0, i2 = 0;

  for (int base = 0; base < M_PTS; base += KNN_TILE) {
    for (int t = threadIdx.x; t < KNN_TILE; t += blockDim.x) {
      sx[t] = pos_sub[(base + t) * 3 + 0];
      sy[t] = pos_sub[(base + t) * 3 + 1];
      sz[t] = pos_sub[(base + t) * 3 + 2];
    }
    __syncthreads();
    for (int t = 0; t < KNN_TILE; ++t) {
      float dx = px - sx[t];
      float dy = py - sy[t];
      float dz = pz - sz[t];
      float d = dx * dx + dy * dy + dz * dz;
      int j = base + t;
      if (d < d2v) {
        if (d < d1) {
          d2v = d1; i2 = i1;
          if (d < d0) { d1 = d0; i1 = i0; d0 = d; i0 = j; }
          else        { d1 = d;  i1 = j; }
        } else {
          d2v = d; i2 = j;
        }
      }
    }
    __syncthreads();
  }

  float w0 = 1.f / fmaxf(d0,  1e-16f);
  float w1 = 1.f / fmaxf(d1,  1e-16f);
  float w2 = 1.f / fmaxf(d2v, 1e-16f);
  float inv = 1.f / (w0 + w1 + w2);
  ow[i * 3 + 0] = w0 * inv;
  ow[i * 3 + 1] = w1 * inv;
  ow[i * 3 + 2] = w2 * inv;
  oidx[i * 3 + 0] = i0;
  oidx[i * 3 + 1] = i1;
  oidx[i * 3 + 2] = i2;
}

// ---------------------------------------------------------------------------
// out[i,:] = relu(BN2(H2[i,:])) + sum_k w[i,k] * feat[idx[i,k], :]
// One block per row; 128 threads -> coalesced row gathers.
// ---------------------------------------------------------------------------
__global__ __launch_bounds__(128) void final_combine(
    const float* __restrict__ H2, const float* __restrict__ mean,
    const float* __restrict__ istd, const float* __restrict__ g,
    const float* __restrict__ be, const float* __restrict__ feat,
    const int* __restrict__ kidx, const float* __restrict__ kw,
    float* __restrict__ out)
{
  const int i = blockIdx.x;
  const int c = threadIdx.x;
  const int j0 = kidx[i * 3 + 0];
  const int j1 = kidx[i * 3 + 1];
  const int j2 = kidx[i * 3 + 2];
  const float w0 = kw[i * 3 + 0];
  const float w1 = kw[i * 3 + 1];
  const float w2 = kw[i * 3 + 2];

  float v = (H2[(size_t)i * C_OUT + c] - mean[c]) * istd[c] * g[c] + be[c];
  v = fmaxf(v, 0.f);
  v += w0 * feat[(size_t)j0 * C_OUT + c]
     + w1 * feat[(size_t)j1 * C_OUT + c]
     + w2 * feat[(size_t)j2 * C_OUT + c];
  out[(size_t)i * C_OUT + c] = v;
}

// ---------------------------------------------------------------------------
extern "C" void kernel_launch(void* const* d_in, const int* in_sizes, int n_in,
                              void* d_out, int out_size, void* d_ws, size_t ws_size,
                              hipStream_t stream) {
  const float* x       = (const float*)d_in[0];   // [16384,128]
  const float* x_sub   = (const float*)d_in[1];   // [8192,256]
  const float* pos     = (const float*)d_in[2];   // [16384,3]
  const float* pos_sub = (const float*)d_in[3];   // [8192,3]
  const float* W_sub   = (const float*)d_in[4];   // [256,128]
  const float* b_sub   = (const float*)d_in[5];   // [128]
  const float* g_sub   = (const float*)d_in[6];
  const float* be_sub  = (const float*)d_in[7];
  const float* Wm      = (const float*)d_in[8];   // [128,128]
  const float* b       = (const float*)d_in[9];
  const float* g       = (const float*)d_in[10];
  const float* be      = (const float*)d_in[11];
  float* out = (float*)d_out;

  char* ws = (char*)d_ws;
  float* h_sub = (float*)ws;                                         // 8192*128 f32
  float* H2    = (float*)(ws + (size_t)M_PTS * C_OUT * 4);           // 16384*128 f32
  float* stats = (float*)(ws + (size_t)(M_PTS + N_PTS) * C_OUT * 4); // 4*128 f32
  float* mean1 = stats;
  float* istd1 = stats + 128;
  float* mean2 = stats + 256;
  float* istd2 = stats + 384;
  int*   kidx  = (int*)(stats + 512);                                // 16384*3 i32
  float* kw    = (float*)(kidx + (size_t)N_PTS * 3);                 // 16384*3 f32

  // MLP1 linear: h_sub = x_sub @ W_sub + b_sub
  gemm_bias_wmma<<<(M_PTS / 16 * 2) / 8, 256, 0, stream>>>(x_sub, W_sub, b_sub, h_sub, M_PTS, C_IN);
  // BN1 stats + apply + ReLU -> feat_sub (in place)
  bn_stats<<<C_OUT, 256, 0, stream>>>(h_sub, mean1, istd1, M_PTS);
  bn_apply_relu<<<(M_PTS * C_OUT) / 256, 256, 0, stream>>>(h_sub, mean1, istd1, g_sub, be_sub);
  // MLP2 linear: H2 = x @ W + b
  gemm_bias_wmma<<<(N_PTS / 16 * 2) / 8, 256, 0, stream>>>(x, Wm, b, H2, N_PTS, C_OUT);
  bn_stats<<<C_OUT, 256, 0, stream>>>(H2, mean2, istd2, N_PTS);
  // kNN (independent of the MLPs)
  knn_top3<<<N_PTS / 256, 256, 0, stream>>>(pos, pos_sub, kidx, kw);
  // out = relu(BN2(H2)) + knn-interp(feat_sub)
  final_combine<<<N_PTS, C_OUT, 0, stream>>>(H2, mean2, istd2, g, be, h_sub, kidx, kw, out);
}